// TopKStrategy_15350213116216
// MI455X (gfx1250) — compile-verified
//
#include <hip/hip_runtime.h>
#include <cstdint>

#define TOPK 64
#define COLS 2048
#define TPB  256
#define EPT  8                 // elements per thread: 2 x float4
#define NWAVE (TPB / 32)       // 8 wave32s per workgroup

typedef float v4f __attribute__((ext_vector_type(4)));

// Order-preserving float -> uint32 map (larger float => larger key).
__device__ __forceinline__ unsigned f2ord(float x) {
  unsigned u = __float_as_uint(x);
  return (u & 0x80000000u) ? ~u : (u | 0x80000000u);
}

// ds_swizzle_b32 XOR butterfly on a float. Group-of-32 encoding:
// offset[14:10]=xor_mask, [9:5]=or_mask(0), [4:0]=and_mask(0x1F).
#define SWZ_XOR_F(x, mask)                                                     \
  __uint_as_float((unsigned)__builtin_amdgcn_ds_swizzle(                       \
      (int)__float_as_uint(x), ((mask) << 10) | 0x1F))

__device__ __forceinline__ float wave_max_f(float x) {
  x = fmaxf(x, SWZ_XOR_F(x, 16));
  x = fmaxf(x, SWZ_XOR_F(x, 8));
  x = fmaxf(x, SWZ_XOR_F(x, 4));
  x = fmaxf(x, SWZ_XOR_F(x, 2));
  x = fmaxf(x, SWZ_XOR_F(x, 1));
  return x;
}

__device__ __forceinline__ float wave_sum_f(float x) {
  x += SWZ_XOR_F(x, 16);
  x += SWZ_XOR_F(x, 8);
  x += SWZ_XOR_F(x, 4);
  x += SWZ_XOR_F(x, 2);
  x += SWZ_XOR_F(x, 1);
  return x;
}

__global__ __launch_bounds__(TPB) void topk_softmax_kernel(const float* __restrict__ in,
                                                           float* __restrict__ out) {
  __shared__ float    lds_row[COLS];     // 8 KB staged row
  __shared__ unsigned hist[TPB];         // 256 bins
  __shared__ unsigned waveTot[NWAVE];    // per-wave histogram totals
  __shared__ float    wred[NWAVE];       // per-wave float reductions
  __shared__ unsigned sel_digit, sel_above;

  const unsigned tid  = threadIdx.x;
  const unsigned lane = tid & 31u;
  const unsigned wv   = tid >> 5;
  const unsigned long long row = blockIdx.x;
  const float* rowp = in + row * (unsigned long long)COLS;

  // ---- CDNA5 async copy global -> LDS (ASYNCcnt), non-temporal: the row is
  // read exactly once, keep it out of WGP$/L2.
  unsigned lds_addr = (unsigned)(size_t)(&lds_row[0]) + tid * 16u;
  unsigned long long gaddr = (unsigned long long)(size_t)rowp + (unsigned long long)tid * 16ull;
  // INST_OFFSET applies to BOTH the LDS and global address (ISA 08 s4.4).
  asm volatile("global_load_async_to_lds_b128 %0, %1, off th:TH_LOAD_NT"
               :: "v"(lds_addr), "v"(gaddr) : "memory");
  asm volatile("global_load_async_to_lds_b128 %0, %1, off offset:4096 th:TH_LOAD_NT"
               :: "v"(lds_addr), "v"(gaddr) : "memory");
  asm volatile("s_wait_asynccnt 0x0" ::: "memory");
  __syncthreads();

  // ---- pull this thread's 8 elements into registers ------------------------
  float v[EPT];
  {
    const float4* l4 = (const float4*)lds_row;
    float4 a = l4[tid];
    float4 b = l4[TPB + tid];
    v[0] = a.x; v[1] = a.y; v[2] = a.z; v[3] = a.w;
    v[4] = b.x; v[5] = b.y; v[6] = b.z; v[7] = b.w;
  }
  unsigned k[EPT];
#pragma unroll
  for (int i = 0; i < EPT; ++i) k[i] = f2ord(v[i]);

  // ---- row max: swizzle butterfly + cross-wave combine (1 barrier) ---------
  float m = v[0];
#pragma unroll
  for (int i = 1; i < EPT; ++i) m = fmaxf(m, v[i]);
  m = wave_max_f(m);
  if (lane == 0) wred[wv] = m;
  __syncthreads();
  float rowmax = wred[0];
#pragma unroll
  for (int w2 = 1; w2 < NWAVE; ++w2) rowmax = fmaxf(rowmax, wred[w2]);

  // ---- exact 64th-largest: 4-pass radix select, wave-level suffix scans ----
  unsigned prefMask = 0u, prefVal = 0u, want = TOPK;
  for (int shift = 24; shift >= 0; shift -= 8) {
    hist[tid] = 0u;
    __syncthreads();
#pragma unroll
    for (int i = 0; i < EPT; ++i)
      if ((k[i] & prefMask) == prefVal)
        atomicAdd(&hist[(k[i] >> shift) & 0xFFu], 1u);
    __syncthreads();

    // wave32 suffix scan of this wave's 32 bins (no barriers)
    const unsigned h = hist[tid];
    unsigned s = h;
#pragma unroll
    for (int off = 1; off < 32; off <<= 1) {
      unsigned t = __shfl_down(s, off, 32);
      if (lane + off < 32) s += t;
    }
    if (lane == 0) waveTot[wv] = s;     // lane 0 holds the wave's bin total
    __syncthreads();

    unsigned sfx = s;                   // suffix over ALL 256 bins
#pragma unroll
    for (int w2 = 0; w2 < NWAVE; ++w2)
      if (w2 > wv) sfx += waveTot[w2];
    const unsigned sfx1 = sfx - h;      // suffix starting at bin tid+1

    if (sfx >= want && sfx1 < want) { sel_digit = tid; sel_above = sfx1; }
    __syncthreads();
    prefVal  |= sel_digit << shift;
    prefMask |= 0xFFu << shift;
    want     -= sel_above;
    // no trailing barrier needed: next pass's two barriers order hist/sel reuse
  }
  const unsigned thrKey = prefVal;      // exact key of the 64th-largest value

  // ---- masked exp + row sum (swizzle butterfly + 1 barrier) ----------------
  float e[EPT];
  float p = 0.0f;
#pragma unroll
  for (int i = 0; i < EPT; ++i) {
    e[i] = (k[i] >= thrKey) ? __expf(v[i] - rowmax) : 0.0f;
    p += e[i];
  }
  p = wave_sum_f(p);
  if (lane == 0) wred[wv] = p;
  __syncthreads();
  float tot = 0.0f;
#pragma unroll
  for (int w2 = 0; w2 < NWAVE; ++w2) tot += wred[w2];
  const float inv = 1.0f / tot;

  // ---- non-temporal streamed store (written once, never re-read) -----------
  float* orow = out + row * (unsigned long long)COLS;
  v4f o0 = { e[0] * inv, e[1] * inv, e[2] * inv, e[3] * inv };
  v4f o1 = { e[4] * inv, e[5] * inv, e[6] * inv, e[7] * inv };
  __builtin_nontemporal_store(o0, (v4f*)orow + tid);
  __builtin_nontemporal_store(o1, (v4f*)orow + TPB + tid);
}

extern "C" void kernel_launch(void* const* d_in, const int* in_sizes, int n_in,
                              void* d_out, int out_size, void* d_ws, size_t ws_size,
                              hipStream_t stream) {
  (void)n_in; (void)d_ws; (void)ws_size; (void)out_size;
  const float* in = (const float*)d_in[0];
  float* out = (float*)d_out;
  const long long total = (long long)in_sizes[0];
  const int rows = (int)(total / COLS);   // 65536 for (2,16,2048,2048)
  topk_softmax_kernel<<<rows, TPB, 0, stream>>>(in, out);
}